// CTCLoss_65816078844676
// MI455X (gfx1250) — compile-verified
//
#include <hip/hip_runtime.h>
#include <stdint.h>

// Problem constants (match reference)
#define TT 256
#define BB 64
#define CC 6000
#define SS 32
#define LL 65            // 2*S+1
#define CHUNK 64
#define NCHUNK (TT / CHUNK)

typedef __attribute__((ext_vector_type(2))) float v2f;
typedef __attribute__((ext_vector_type(8))) float v8f;

// One workgroup per batch element.
// Wave 0: sequential alpha recursion (double-buffered in LDS).
// Waves 1..7: async-prefetch next 64-timestep chunk of gathered log-probs
//             into LDS via global_load_async_to_lds_b32 (ASYNCcnt).
__global__ __launch_bounds__(256) void ctc_alpha_kernel(
    const float* __restrict__ logp,     // (T, B, C)
    const int*   __restrict__ targets,  // (B, S)
    const int*   __restrict__ ilen,     // (B,)
    const int*   __restrict__ tlen,     // (B,)
    float*       __restrict__ loss_ws)  // (B,) per-batch loss / target_length
{
    __shared__ float s_lp[2][CHUNK * LL];   // raw log-probs (exp applied in DP)
    __shared__ int   s_tp[LL];              // extended targets
    __shared__ float s_mask[LL];            // skip-transition mask (0/1)
    __shared__ float s_alpha[2][LL + 2];    // alpha ping-pong, 2-cell zero guard

    const int b    = blockIdx.x;
    const int tid  = threadIdx.x;
    const int lane = tid & 31;
    const int wid  = tid >> 5;

    const int il = ilen[b];
    const int tl = tlen[b];

    // Extended target sequence: blank at even s, label at odd s (< 2*tl)
    if (tid < LL) {
        int v = 0;
        if (tid & 1) {
            int kk = tid >> 1;
            if (kk < tl) v = targets[b * SS + kk];
        }
        s_tp[tid] = v;
    }
    if (tid < 2 * (LL + 2)) ((float*)s_alpha)[tid] = 0.0f;
    __syncthreads();

    if (tid < LL)
        s_mask[tid] = (tid >= 2 && s_tp[tid] != s_tp[tid - 2]) ? 1.0f : 0.0f;

    // ---- chunk 0 gather: all 256 threads, async global->LDS ----
    for (int idx = tid; idx < CHUNK * LL; idx += 256) {
        int tloc = idx / LL;
        int s    = idx - tloc * LL;
        size_t   gi = ((size_t)tloc * BB + b) * (size_t)CC + (size_t)s_tp[s];
        uint64_t ga = (uint64_t)(uintptr_t)(logp + gi);
        uint32_t la = (uint32_t)(uintptr_t)&s_lp[0][idx];
        asm volatile("global_load_async_to_lds_b32 %0, %1, off"
                     :: "v"(la), "v"(ga) : "memory");
    }
    asm volatile("s_wait_asynccnt 0x0" ::: "memory");
    __syncthreads();

    // per-lane DP constants (wave 0 only)
    float m0 = 0.0f, m1 = 0.0f, m2 = 0.0f;
    if (wid == 0) {
        m0 = s_mask[lane];
        m1 = s_mask[32 + lane];
        m2 = (lane == 0) ? s_mask[64] : 0.0f;
    }

    int cur = 0;
    if (wid == 0 && lane == 0) {
        // alpha0: positions 0 and 1 from t=0 emissions
        s_alpha[0][2 + 0] = __expf(s_lp[0][0]);
        s_alpha[0][2 + 1] = __expf(s_lp[0][1]);
    }

    for (int k = 0; k < NCHUNK; ++k) {
        if (wid != 0) {
            // -------- producers: prefetch chunk k+1 --------
            if (k + 1 < NCHUNK) {
                int buf = (k + 1) & 1;
                for (int idx = tid - 32; idx < CHUNK * LL; idx += 224) {
                    int tloc = idx / LL;
                    int s    = idx - tloc * LL;
                    int t    = (k + 1) * CHUNK + tloc;
                    size_t   gi = ((size_t)t * BB + b) * (size_t)CC + (size_t)s_tp[s];
                    uint64_t ga = (uint64_t)(uintptr_t)(logp + gi);
                    uint32_t la = (uint32_t)(uintptr_t)&s_lp[buf][idx];
                    asm volatile("global_load_async_to_lds_b32 %0, %1, off"
                                 :: "v"(la), "v"(ga) : "memory");
                }
                asm volatile("s_wait_asynccnt 0x0" ::: "memory");
            }
        } else {
            // -------- consumer: DP over this chunk (single wave, lockstep) --------
            const float* lpbuf = s_lp[k & 1];
            const int t0 = (k == 0) ? 1 : k * CHUNK;
            const int t1 = (k + 1) * CHUNK;
            for (int t = t0; t < t1; ++t) {
                float* A0 = s_alpha[cur];       // read
                float* A1 = s_alpha[cur ^ 1];   // write
                const int  tloc = t - k * CHUNK;
                const bool act  = (t < il);
                {   // s = lane (0..31)
                    int s = lane;
                    float a0 = A0[2 + s], a1 = A0[1 + s], a2 = A0[s];
                    float p  = __expf(lpbuf[tloc * LL + s]);
                    float nv = p * (a0 + a1 + m0 * a2);
                    A1[2 + s] = act ? nv : a0;
                }
                {   // s = 32+lane (32..63)
                    int s = 32 + lane;
                    float a0 = A0[2 + s], a1 = A0[1 + s], a2 = A0[s];
                    float p  = __expf(lpbuf[tloc * LL + s]);
                    float nv = p * (a0 + a1 + m1 * a2);
                    A1[2 + s] = act ? nv : a0;
                }
                if (lane == 0) {  // s = 64
                    int s = 64;
                    float a0 = A0[2 + s], a1 = A0[1 + s], a2 = A0[s];
                    float p  = __expf(lpbuf[tloc * LL + s]);
                    float nv = p * (a0 + a1 + m2 * a2);
                    A1[2 + s] = act ? nv : a0;
                }
                cur ^= 1;
            }
        }
        __syncthreads();
    }

    if (wid == 0 && lane == 0) {
        const float* A0 = s_alpha[cur];
        int   e = 2 * tl;
        float a = A0[2 + e] + A0[1 + e];       // alpha[end] + alpha[end-1]
        loss_ws[b] = -logf(a) / (float)tl;
    }
}

// Final mean over 64 per-batch losses via V_WMMA_F32_16X16X4_F32.
// B operand (4x16) holds the 64 values; A = ones => every row of D is the
// vector of column sums. d[0] across the 32 lanes holds each column sum
// exactly twice (rows M=0 and M=8), so wave-reduce(d[0]) = 2 * sum.
__global__ __launch_bounds__(32) void ctc_reduce_kernel(
    const float* __restrict__ v, float* __restrict__ out)
{
    const int l  = threadIdx.x;
    const int n  = l & 15;
    const int hi = l >> 4;   // 0: rows K=0/1, 1: rows K=2/3

    v2f a; a.x = 1.0f; a.y = 1.0f;
    v2f bm;
    bm.x = v[(hi ? 2 : 0) * 16 + n];
    bm.y = v[(hi ? 3 : 1) * 16 + n];
    v8f c = {};
    v8f d = __builtin_amdgcn_wmma_f32_16x16x4_f32(
        /*neg_a=*/false, a, /*neg_b=*/false, bm,
        /*c_mod=*/(short)0, c, /*reuse_a=*/false, /*reuse_b=*/false);

    float t = d[0];
    for (int m = 16; m > 0; m >>= 1) t += __shfl_xor(t, m, 32);
    if (l == 0) out[0] = t * (1.0f / 128.0f);  // /2 (dup) /64 (mean)
}

extern "C" void kernel_launch(void* const* d_in, const int* in_sizes, int n_in,
                              void* d_out, int out_size, void* d_ws, size_t ws_size,
                              hipStream_t stream)
{
    (void)in_sizes; (void)n_in; (void)out_size; (void)ws_size;
    const float* logp    = (const float*)d_in[0];
    const int*   targets = (const int*)d_in[1];
    const int*   ilens   = (const int*)d_in[2];
    const int*   tlens   = (const int*)d_in[3];
    float* ws  = (float*)d_ws;
    float* out = (float*)d_out;

    ctc_alpha_kernel<<<BB, 256, 0, stream>>>(logp, targets, ilens, tlens, ws);
    ctc_reduce_kernel<<<1, 32, 0, stream>>>(ws, out);
}